// Point_AVGPool_59296318489054
// MI455X (gfx1250) — compile-verified
//
#include <hip/hip_runtime.h>

typedef __attribute__((ext_vector_type(2))) float v2f;
typedef __attribute__((ext_vector_type(8))) float v8f;

#define RADIUS_F   0.2f
#define NSAMPLE    16
#define BATCH      4

// ----------------------------- FPS ---------------------------------
// One workgroup per batch. Min-distances held in registers (8 pts/thread).
// Per-step argmax: wave32 shfl_xor butterfly (in-register), then a single
// 32-value LDS stage -> 2 barriers/step instead of 10.
#define FPS_THREADS 1024
#define FPS_PTS     8
#define FPS_WAVES   (FPS_THREADS / 32)

__global__ __launch_bounds__(FPS_THREADS)
void fps_kernel(const float* __restrict__ xyz, int* __restrict__ inds,
                int N, int npoint)
{
    const int b   = blockIdx.x;
    const int tid = threadIdx.x;
    const float* base = xyz + (size_t)b * N * 3;

    __shared__ float swval[FPS_WAVES];
    __shared__ int   swidx[FPS_WAVES];
    __shared__ int   slast;

    float px[FPS_PTS], py[FPS_PTS], pz[FPS_PTS], md[FPS_PTS];
#pragma unroll
    for (int k = 0; k < FPS_PTS; ++k) {
        int i = tid + k * FPS_THREADS;
        if (i >= N) i = N - 1;
        px[k] = base[i*3+0]; py[k] = base[i*3+1]; pz[k] = base[i*3+2];
        md[k] = 1e10f;
    }
    if (tid == 0) { slast = 0; inds[(size_t)b * npoint] = 0; }
    __syncthreads();

    const int wave = tid >> 5;
    const int wl   = tid & 31;

    for (int it = 1; it < npoint; ++it) {
        const int last = slast;
        const float lx = base[last*3+0];
        const float ly = base[last*3+1];
        const float lz = base[last*3+2];

        float best = -1.0f; int bidx = tid;
#pragma unroll
        for (int k = 0; k < FPS_PTS; ++k) {
            float dx = px[k]-lx, dy = py[k]-ly, dz = pz[k]-lz;
            float d  = dx*dx + dy*dy + dz*dz;
            md[k] = fminf(md[k], d);
            if (md[k] > best) { best = md[k]; bidx = tid + k * FPS_THREADS; }
        }

        // wave-local argmax (value, index) butterfly
#pragma unroll
        for (int off = 16; off > 0; off >>= 1) {
            float v2 = __shfl_xor(best, off, 32);
            int   i2 = __shfl_xor(bidx, off, 32);
            if (v2 > best || (v2 == best && i2 < bidx)) { best = v2; bidx = i2; }
        }
        if (wl == 0) { swval[wave] = best; swidx[wave] = bidx; }
        __syncthreads();

        if (tid < 32) {   // exactly wave 0, full EXEC
            best = swval[tid]; bidx = swidx[tid];
#pragma unroll
            for (int off = 16; off > 0; off >>= 1) {
                float v2 = __shfl_xor(best, off, 32);
                int   i2 = __shfl_xor(bidx, off, 32);
                if (v2 > best || (v2 == best && i2 < bidx)) { best = v2; bidx = i2; }
            }
            if (tid == 0) { slast = bidx; inds[(size_t)b * npoint + it] = bidx; }
        }
        __syncthreads();
    }
}

// ----------------------- gather selected xyz ------------------------
__global__ void gather_xyz_kernel(const float* __restrict__ xyz,
                                  const int* __restrict__ inds,
                                  float* __restrict__ out_xyz, int N, int S)
{
    int t = blockIdx.x * blockDim.x + threadIdx.x;
    if (t >= BATCH * S) return;
    int b  = t / S;
    int id = inds[t];
    const float* p = xyz + ((size_t)b * N + id) * 3;
    out_xyz[t*3+0] = p[0];
    out_xyz[t*3+1] = p[1];
    out_xyz[t*3+2] = p[2];
}

// --------------------- ball query via f32 WMMA ----------------------
// One wave per 16 centers. Per 16x16 tile:
//   A row m   = ( cx, cy, cz, 1 )          (16x4)
//   B col j   = (-2px, -2py, -2pz, |p|^2)  (4x16)
//   D = A*B   = |p|^2 - 2 c.p              (16x16)
// Hit test:  D < r^2 - |c|^2  (per-center threshold, C = inline 0).
// Fragment construction is branchless (v_cndmask) to keep EXEC full and
// avoid per-tile saveexec overhead around the WMMAs.
#define BQ_CHUNK 128

__global__ __launch_bounds__(32)
void ball_query_wmma_kernel(const float* __restrict__ xyz,
                            const float* __restrict__ centers,
                            int* __restrict__ ballIdx, int N, int S)
{
    __shared__ float d2s[16 * BQ_CHUNK];
    __shared__ int   s_flag;

    const int tilesPerBatch = S / 16;
    const int b    = blockIdx.x / tilesPerBatch;
    const int s0   = (blockIdx.x % tilesPerBatch) * 16;

    const int  lane = threadIdx.x;
    const bool hi   = (lane & 16) != 0;   // lanes 16..31 carry K=2,3
    const int  l15  = lane & 15;

    const float* cen = centers + ((size_t)b * S + s0) * 3;
    const float cx = cen[l15*3+0], cy = cen[l15*3+1], cz = cen[l15*3+2];
    const float thr = RADIUS_F * RADIUS_F - (cx*cx + cy*cy + cz*cz);

    v2f A;
    A.x = hi ? cz   : cx;
    A.y = hi ? 1.0f : cy;

    const float* pbase = xyz + (size_t)b * N * 3;
    const v8f czero = {};
    const int rowOff = hi ? 8 : 0;

    int  cnt = 0, first = 0;
    int* myIdx = ballIdx + ((size_t)b * S + s0 + l15) * NSAMPLE;

    for (int c0 = 0; c0 < N; c0 += BQ_CHUNK) {
        // prefetch next chunk of points while we compute + scan this one
        if (c0 + BQ_CHUNK < N)
            __builtin_prefetch(pbase + (size_t)(c0 + BQ_CHUNK + l15*8) * 3, 0, 0);

        // 16 x BQ_CHUNK partial-d2 tile into LDS: 8 WMMAs
#pragma unroll
        for (int t = 0; t < BQ_CHUNK / 16; ++t) {
            int j = c0 + t*16 + l15;
            float qx = pbase[j*3+0], qy = pbase[j*3+1], qz = pbase[j*3+2];
            v2f Bf;
            Bf.x = -2.0f * (hi ? qz : qx);
            Bf.y = hi ? (qx*qx + qy*qy + qz*qz) : (-2.0f * qy);

            v8f d = __builtin_amdgcn_wmma_f32_16x16x4_f32(
                        false, A, false, Bf, (short)0, czero, false, false);
#pragma unroll
            for (int v = 0; v < 8; ++v)
                d2s[(v + rowOff) * BQ_CHUNK + t*16 + l15] = d[v];
        }
        asm volatile("s_wait_dscnt 0" ::: "memory");   // cross-lane LDS visibility

        // in-order "first NSAMPLE hits" scan; lane m owns center s0+m
        if (lane < 16) {
            for (int col = 0; col < BQ_CHUNK && cnt < NSAMPLE; ++col) {
                if (d2s[lane * BQ_CHUNK + col] < thr) {
                    int gi = c0 + col;
                    if (cnt == 0) first = gi;
                    myIdx[cnt] = gi;
                    ++cnt;
                }
            }
        }
        // wave-uniform early exit via LDS flag (same-wave DS ops are in-order)
        if (lane == 0) s_flag = 0;
        asm volatile("s_wait_dscnt 0" ::: "memory");
        if (lane < 16 && cnt < NSAMPLE) s_flag = 1;
        asm volatile("s_wait_dscnt 0" ::: "memory");
        if (s_flag == 0) break;
    }

    if (lane < 16) {
        int pad = (cnt > 0) ? first : 0;
        for (int k = cnt; k < NSAMPLE; ++k) myIdx[k] = pad;
    }
}

// -------------------- group features + mean pool --------------------
// One wave per (b,s) group; 32 lanes x float4 = one 128-float row / load.
__global__ __launch_bounds__(256)
void pool_kernel(const float* __restrict__ feat,
                 const int* __restrict__ ballIdx,
                 float* __restrict__ outPool, int N, int S)
{
    int gt   = blockIdx.x * blockDim.x + threadIdx.x;
    int w    = gt >> 5;
    int lane = gt & 31;
    if (w >= BATCH * S) return;
    int b = w / S;

    const int* gi = ballIdx + (size_t)w * NSAMPLE;
    float a0 = 0.f, a1 = 0.f, a2 = 0.f, a3 = 0.f;
#pragma unroll
    for (int k = 0; k < NSAMPLE; ++k) {
        int id = gi[k];
        const float4* row = (const float4*)(feat + ((size_t)b * N + id) * 128);
        float4 v = row[lane];
        a0 += v.x; a1 += v.y; a2 += v.z; a3 += v.w;
    }
    const float inv = 1.0f / (float)NSAMPLE;
    float4 o; o.x = a0*inv; o.y = a1*inv; o.z = a2*inv; o.w = a3*inv;
    ((float4*)(outPool + (size_t)w * 128))[lane] = o;
}

// ------------------------------ launch ------------------------------
extern "C" void kernel_launch(void* const* d_in, const int* in_sizes, int n_in,
                              void* d_out, int out_size, void* d_ws, size_t ws_size,
                              hipStream_t stream)
{
    const float* points   = (const float*)d_in[0];   // [4, N, 3]
    const float* features = (const float*)d_in[1];   // [4, N, 128]
    float* out = (float*)d_out;

    const int B = BATCH;
    const int N = in_sizes[0] / (3 * B);   // 8192
    const int S = (N + 3) / 4;             // 2048

    int* inds    = (int*)d_ws;                                          // B*S
    int* ballIdx = (int*)((char*)d_ws + (size_t)B * S * sizeof(int));   // B*S*16

    float* out_xyz  = out;                       // B*S*3
    float* out_pool = out + (size_t)B * S * 3;   // B*S*128

    fps_kernel<<<B, FPS_THREADS, 0, stream>>>(points, inds, N, S);
    gather_xyz_kernel<<<(B * S + 255) / 256, 256, 0, stream>>>(points, inds, out_xyz, N, S);
    ball_query_wmma_kernel<<<B * (S / 16), 32, 0, stream>>>(points, out_xyz, ballIdx, N, S);
    pool_kernel<<<(B * S * 32 + 255) / 256, 256, 0, stream>>>(features, ballIdx, out_pool, N, S);
}